// Performer_9242769621383
// MI455X (gfx1250) — compile-verified
//
#include <hip/hip_runtime.h>
#include <math.h>

// Performer causal linear attention, chunked formulation for MI455X (gfx1250).
// wave32, f32 WMMA 16x16x4, 256-thread blocks (8 waves), LDS stride-68 padding.

typedef __attribute__((ext_vector_type(2))) float v2f;
typedef __attribute__((ext_vector_type(8))) float v8f;

#define LDSS 68   // LDS row stride in floats (64 + 4 pad -> conflict-free frag loads)
#define CK   64   // chunk length
#define DM   64   // D == M == 64

// ---- WMMA fragment loaders (ISA 7.12.2 layouts) ----------------------------
// A 16x4 (f32): lane 0-15 hold rows M=0..15; v0 = K=(0|2), v1 = K=(1|3) by lane half.
__device__ __forceinline__ v2f ld_a(const float* __restrict__ A, int lda,
                                    int row, int kb, int lane) {
  int r = row + (lane & 15);
  int k = kb + ((lane >> 4) << 1);
  v2f a; a.x = A[r * lda + k]; a.y = A[r * lda + k + 1]; return a;
}
// A fragment where logical A[m][k] = S[k][m] (transposed source in LDS)
__device__ __forceinline__ v2f ld_at(const float* __restrict__ S, int lds_,
                                     int row, int kb, int lane) {
  int m = row + (lane & 15);
  int k = kb + ((lane >> 4) << 1);
  v2f a; a.x = S[k * lds_ + m]; a.y = S[(k + 1) * lds_ + m]; return a;
}
// B 4x16 (f32): row-per-VGPR striped over lanes; K split by lane half.
__device__ __forceinline__ v2f ld_b(const float* __restrict__ Bm, int ldb,
                                    int kb, int col, int lane) {
  int n = col + (lane & 15);
  int k = kb + ((lane >> 4) << 1);
  v2f b; b.x = Bm[k * ldb + n]; b.y = Bm[(k + 1) * ldb + n]; return b;
}
// B fragment where logical B[k][n] = S[n][k]
__device__ __forceinline__ v2f ld_bt(const float* __restrict__ S, int lds_,
                                     int kb, int col, int lane) {
  int n = col + (lane & 15);
  int k = kb + ((lane >> 4) << 1);
  v2f b; b.x = S[n * lds_ + k]; b.y = S[n * lds_ + k + 1]; return b;
}

// One 16x16 output tile of a 64-K GEMM, accumulating into acc.
template <bool AT, bool BT>
__device__ __forceinline__ v8f gemm64(const float* __restrict__ A, int lda,
                                      const float* __restrict__ Bm, int ldb,
                                      int row, int col, int lane, v8f acc) {
#pragma unroll
  for (int k = 0; k < DM; k += 4) {
    v2f a, b;
    if constexpr (AT) a = ld_at(A, lda, row, k, lane); else a = ld_a(A, lda, row, k, lane);
    if constexpr (BT) b = ld_bt(Bm, ldb, k, col, lane); else b = ld_b(Bm, ldb, k, col, lane);
    acc = __builtin_amdgcn_wmma_f32_16x16x4_f32(false, a, false, b,
                                                (short)0, acc, false, false);
  }
  return acc;
}

// Cooperative 64x64 f32 tile load: global (row-major, contiguous) -> LDS (stride 68).
__device__ __forceinline__ void load_chunk64(float* dst, const float* __restrict__ src,
                                             int tid) {
  for (int i = tid; i < (CK * DM) / 4; i += 256) {
    int r = i >> 4;
    int c = (i & 15) << 2;
    const float4 v = *reinterpret_cast<const float4*>(src + r * DM + c);
    dst[r * LDSS + c + 0] = v.x;
    dst[r * LDSS + c + 1] = v.y;
    dst[r * LDSS + c + 2] = v.z;
    dst[r * LDSS + c + 3] = v.w;
  }
}

// projT[d][m] = proj[m][d] so that  logits = X @ projT  is a plain row-major GEMM.
__device__ __forceinline__ void load_projT(float* dst, const float* __restrict__ proj,
                                           int tid) {
  for (int i = tid; i < DM * DM; i += 256) {
    int m = i >> 6, d = i & 63;
    dst[d * LDSS + m] = proj[m * DM + d];
  }
}

// ---- Kernel 1: per-timestep scales (cross-batch Frobenius norm) ------------
// scale[l] = exp(-0.5*sqrt(sum_{b,d} X[b,l,d]^2)) / sqrt(M)   (1/sqrt(M) folded in)
__global__ void __launch_bounds__(256) perf_scales(const float* __restrict__ Q,
                                                   const float* __restrict__ Kin,
                                                   float* __restrict__ scaleQ,
                                                   float* __restrict__ scaleK,
                                                   int Bn, int L) {
  const int lane = threadIdx.x & 31, wave = threadIdx.x >> 5;
  const int l = blockIdx.x * 8 + wave;
  if (l >= L) return;
  const float* X = blockIdx.y ? Kin : Q;
  float* S = blockIdx.y ? scaleK : scaleQ;
  float s = 0.f;
  for (int b = 0; b < Bn; ++b) {
    const float* p = X + ((size_t)b * L + l) * DM;
    float x0 = p[lane], x1 = p[lane + 32];
    s = fmaf(x0, x0, s);
    s = fmaf(x1, x1, s);
  }
#pragma unroll
  for (int off = 16; off; off >>= 1) s += __shfl_xor(s, off, 32);
  if (lane == 0) S[l] = expf(-0.5f * sqrtf(s)) * 0.125f;  // 1/sqrt(64)
}

// ---- Kernel 2: per-chunk local state  KVc = phiK^T @ Vc, Nc = colsum(phiK) --
__global__ void __launch_bounds__(256) perf_chunk_state(const float* __restrict__ Kin,
                                                        const float* __restrict__ Vin,
                                                        const float* __restrict__ proj,
                                                        const float* __restrict__ scaleK,
                                                        float* __restrict__ state,
                                                        int L) {
  extern __shared__ float smem[];
  float* projT = smem;                       // 64 x LDSS
  float* bufK  = projT + DM * LDSS;          // 64 x LDSS
  float* bufV  = bufK + CK * LDSS;           // 64 x LDSS
  float* phiK  = bufV + CK * LDSS;           // 64 x LDSS

  const int tid = threadIdx.x, lane = tid & 31, wave = tid >> 5;
  const int NC = L / CK;
  const int b  = blockIdx.x / NC;
  const int ci = blockIdx.x - b * NC;
  const int t0 = ci * CK;
  const size_t base = ((size_t)b * L + t0) * DM;

  load_projT(projT, proj, tid);
  load_chunk64(bufK, Kin + base, tid);
  load_chunk64(bufV, Vin + base, tid);
  __syncthreads();

  // phiK = exp(Kc @ projT) * scaleK[row]
#pragma unroll
  for (int tt = 0; tt < 2; ++tt) {
    int t = wave + tt * 8;
    int row = (t >> 2) << 4, col = (t & 3) << 4;
    v8f acc = {};
    acc = gemm64<false, false>(bufK, LDSS, projT, LDSS, row, col, lane, acc);
    int rb = row + ((lane >> 4) << 3);
    int c = col + (lane & 15);
#pragma unroll
    for (int r = 0; r < 8; ++r)
      phiK[(rb + r) * LDSS + c] = expf(acc[r]) * scaleK[t0 + rb + r];
  }
  __syncthreads();

  const int SE = DM * DM + DM;
  float* st = state + (size_t)blockIdx.x * SE;

  // KVc = phiK^T @ Vc  (M x D), written straight to workspace
#pragma unroll
  for (int tt = 0; tt < 2; ++tt) {
    int t = wave + tt * 8;
    int row = (t >> 2) << 4, col = (t & 3) << 4;
    v8f acc = {};
    acc = gemm64<true, false>(phiK, LDSS, bufV, LDSS, row, col, lane, acc);
    int rb = row + ((lane >> 4) << 3);
    int c = col + (lane & 15);
#pragma unroll
    for (int r = 0; r < 8; ++r)
      st[(rb + r) * DM + c] = acc[r];
  }
  // Nc[m] = sum_i phiK[i][m]
  if (tid < DM) {
    float s = 0.f;
#pragma unroll 8
    for (int i = 0; i < CK; ++i) s += phiK[i * LDSS + tid];
    st[DM * DM + tid] = s;
  }
}

// ---- Kernel 3: in-place exclusive prefix scan over chunks ------------------
// One thread per state element (B * 4160 threads), coalesced, sequential only in c.
__global__ void perf_scan(float* __restrict__ state, int Bn, int NC) {
  const int SE = DM * DM + DM;
  int gid = blockIdx.x * blockDim.x + threadIdx.x;
  if (gid >= Bn * SE) return;
  int b = gid / SE, e = gid - b * SE;
  float run = 0.f;
  size_t idx = (size_t)b * NC * SE + e;
  for (int c = 0; c < NC; ++c, idx += SE) {
    float v = state[idx];
    state[idx] = run;   // exclusive prefix: state at chunk start
    run += v;
  }
}

// ---- Kernel 4: chunk outputs -----------------------------------------------
// A = tril(phiQ @ phiK^T); O = A@V + phiQ@KV0; r = rowsum(A) + phiQ.N0; out = O/r'
__global__ void __launch_bounds__(256) perf_output(const float* __restrict__ Qin,
                                                   const float* __restrict__ Kin,
                                                   const float* __restrict__ Vin,
                                                   const float* __restrict__ proj,
                                                   const float* __restrict__ scaleQ,
                                                   const float* __restrict__ scaleK,
                                                   const float* __restrict__ state,
                                                   float* __restrict__ out, int L) {
  extern __shared__ float smem[];
  float* projT = smem;                  // 64 x LDSS
  float* bufQ  = projT + DM * LDSS;     // Qc, later reused for masked A
  float* bufK  = bufQ + CK * LDSS;
  float* bufV  = bufK + CK * LDSS;
  float* phiQ  = bufV + CK * LDSS;
  float* phiK  = phiQ + CK * LDSS;
  float* KV0   = phiK + CK * LDSS;      // 64 x LDSS running state at chunk start
  float* N0    = KV0 + DM * LDSS;       // 64
  float* rbuf  = N0 + DM;               // 64

  const int tid = threadIdx.x, lane = tid & 31, wave = tid >> 5;
  const int NC = L / CK;
  const int b  = blockIdx.x / NC;
  const int ci = blockIdx.x - b * NC;
  const int t0 = ci * CK;
  const size_t base = ((size_t)b * L + t0) * DM;
  const int SE = DM * DM + DM;
  const float* st = state + (size_t)blockIdx.x * SE;

  load_projT(projT, proj, tid);
  load_chunk64(bufQ, Qin + base, tid);
  load_chunk64(bufK, Kin + base, tid);
  load_chunk64(bufV, Vin + base, tid);
  for (int i = tid; i < (DM * DM) / 4; i += 256) {   // stage KV0
    int r = i >> 4, c = (i & 15) << 2;
    const float4 v = *reinterpret_cast<const float4*>(st + r * DM + c);
    KV0[r * LDSS + c + 0] = v.x;
    KV0[r * LDSS + c + 1] = v.y;
    KV0[r * LDSS + c + 2] = v.z;
    KV0[r * LDSS + c + 3] = v.w;
  }
  if (tid < DM) N0[tid] = st[DM * DM + tid];
  __syncthreads();

  // phiQ, phiK
#pragma unroll
  for (int tt = 0; tt < 2; ++tt) {
    int t = wave + tt * 8;
    int row = (t >> 2) << 4, col = (t & 3) << 4;
    int rb = row + ((lane >> 4) << 3);
    int c = col + (lane & 15);
    v8f aq = {};
    aq = gemm64<false, false>(bufQ, LDSS, projT, LDSS, row, col, lane, aq);
    v8f ak = {};
    ak = gemm64<false, false>(bufK, LDSS, projT, LDSS, row, col, lane, ak);
#pragma unroll
    for (int r = 0; r < 8; ++r) {
      phiQ[(rb + r) * LDSS + c] = expf(aq[r]) * scaleQ[t0 + rb + r];
      phiK[(rb + r) * LDSS + c] = expf(ak[r]) * scaleK[t0 + rb + r];
    }
  }
  __syncthreads();

  // A = tril(phiQ @ phiK^T) into bufQ (Qc no longer needed)
#pragma unroll
  for (int tt = 0; tt < 2; ++tt) {
    int t = wave + tt * 8;
    int row = (t >> 2) << 4, col = (t & 3) << 4;
    v8f acc = {};
    acc = gemm64<false, true>(phiQ, LDSS, phiK, LDSS, row, col, lane, acc);
    int rb = row + ((lane >> 4) << 3);
    int c = col + (lane & 15);
#pragma unroll
    for (int r = 0; r < 8; ++r)
      bufQ[(rb + r) * LDSS + c] = (c <= rb + r) ? acc[r] : 0.f;  // causal, inclusive
  }
  __syncthreads();

  // r[i] = rowsum(A)[i] + phiQ[i] . N0  ;  denom = r + sign(r)*1e-6
  if (tid < CK) {
    float s = 0.f, s2 = 0.f;
#pragma unroll 8
    for (int m = 0; m < DM; ++m) {
      s += bufQ[tid * LDSS + m];
      s2 = fmaf(phiQ[tid * LDSS + m], N0[m], s2);
    }
    float rv = s + s2;
    rbuf[tid] = rv + (rv > 0.f ? 1e-6f : (rv < 0.f ? -1e-6f : 0.f));
  }
  __syncthreads();

  // O = A @ V + phiQ @ KV0 ; divide ; store
#pragma unroll
  for (int tt = 0; tt < 2; ++tt) {
    int t = wave + tt * 8;
    int row = (t >> 2) << 4, col = (t & 3) << 4;
    v8f acc = {};
    acc = gemm64<false, false>(bufQ, LDSS, bufV, LDSS, row, col, lane, acc);
    acc = gemm64<false, false>(phiQ, LDSS, KV0, LDSS, row, col, lane, acc);
    int rb = row + ((lane >> 4) << 3);
    int c = col + (lane & 15);
#pragma unroll
    for (int r = 0; r < 8; ++r)
      out[base + (size_t)(rb + r) * DM + c] = acc[r] / rbuf[rb + r];
  }
}

// ---------------------------------------------------------------------------
extern "C" void kernel_launch(void* const* d_in, const int* in_sizes, int n_in,
                              void* d_out, int out_size, void* d_ws, size_t ws_size,
                              hipStream_t stream) {
  (void)n_in; (void)out_size; (void)ws_size;
  const float* Q    = (const float*)d_in[0];
  const float* K    = (const float*)d_in[1];
  const float* V    = (const float*)d_in[2];
  const float* proj = (const float*)d_in[4];   // (M, D) row-major
  float* out = (float*)d_out;

  const int Bn = in_sizes[3] / DM;             // sent_embed_slice is (B, D)
  const int L  = in_sizes[0] / (Bn * DM);
  const int NC = L / CK;
  const int SE = DM * DM + DM;

  float* scaleQ = (float*)d_ws;
  float* scaleK = scaleQ + L;
  float* state  = scaleK + L;                  // B * NC * SE floats

  // 1) per-timestep scales (cross-batch norm), Q and K in parallel via grid.y
  perf_scales<<<dim3((L + 7) / 8, 2), 256, 0, stream>>>(Q, K, scaleQ, scaleK, Bn, L);

  // 2) per-chunk local KV/N states (fully parallel over B*NC chunks)
  const uint32_t smem2 = (DM + 3 * CK) * LDSS * sizeof(float);           // ~68 KB
  hipFuncSetAttribute((const void*)perf_chunk_state,
                      hipFuncAttributeMaxDynamicSharedMemorySize, (int)smem2);
  perf_chunk_state<<<Bn * NC, 256, smem2, stream>>>(K, V, proj, scaleK, state, L);

  // 3) exclusive prefix scan over chunks (parallel across state elements)
  const int scanN = Bn * SE;
  perf_scan<<<(scanN + 255) / 256, 256, 0, stream>>>(state, Bn, NC);

  // 4) chunk outputs (fully parallel over B*NC chunks)
  const uint32_t smem4 = ((DM + 5 * CK + DM) * LDSS + 2 * DM) * sizeof(float); // ~120 KB
  hipFuncSetAttribute((const void*)perf_output,
                      hipFuncAttributeMaxDynamicSharedMemorySize, (int)smem4);
  perf_output<<<Bn * NC, 256, smem4, stream>>>(Q, K, V, proj, scaleQ, scaleK,
                                               state, out, L);
}